// DenseEquivariantMatrix_35253091565745
// MI455X (gfx1250) — compile-verified
//
#include <hip/hip_runtime.h>

typedef __attribute__((ext_vector_type(2))) float v2f;
typedef __attribute__((ext_vector_type(4))) float v4f;
typedef __attribute__((ext_vector_type(8))) float v8f;
typedef __attribute__((ext_vector_type(4))) unsigned int u32x4;
typedef __attribute__((ext_vector_type(4))) int i32x4;
typedef __attribute__((ext_vector_type(8))) int i32x8;

namespace {
constexpr int kB    = 2048;   // batch
constexpr int kFOut = 32;
constexpr int kFIn  = 32;
constexpr int kNS   = 256;    // n_symm
constexpr int kRowK = kFIn * kNS;   // 8192: x row stride == out row stride

constexpr int kBM   = 128;    // b rows per workgroup
constexpr int kBGN  = 64;     // g columns per workgroup
constexpr int kBH   = 32;     // h-block (K chunk per fi)
constexpr int kXPad = 36;     // padded LDS row stride (dwords) -> conflict-free
constexpr int kPPad = 36;
constexpr int kNChunks = (kNS / kBH) * kFIn;  // 256 K-chunks of 32
}

#if defined(__HIP_DEVICE_COMPILE__) && __has_builtin(__builtin_amdgcn_tensor_load_to_lds) && __has_builtin(__builtin_amdgcn_s_wait_tensorcnt)
#define USE_TDM 1
#else
#define USE_TDM 0
#endif

#if USE_TDM
// Issue one TDM 2D tile load: 32 (contiguous f32) x 128 rows, row stride 8192 f32,
// LDS padding 4 dwords after every 32 dwords (-> 36-dword padded rows).
__device__ __forceinline__ void tdm_load_x_tile(const float* gsrc, unsigned lds_off) {
    unsigned long long ga = (unsigned long long)(uintptr_t)gsrc;
    u32x4 g0;
    g0.x = 1u;                                   // count=1, user mode, no gather
    g0.y = lds_off;                              // lds_addr (bytes)
    g0.z = (unsigned)(ga & 0xFFFFFFFFu);         // global_addr[31:0]
    g0.w = (unsigned)((ga >> 32) & 0x01FFFFFFu)  // global_addr[56:32]
         | (2u << 30);                           // type=2 ("image")
    i32x8 g1;
    g1[0] = (2 << 16)        // data_size = 4 bytes
          | (1 << 20)        // pad_enable
          | (4 << 22)        // pad_interval: 2^(4+1)=32 dwords
          | (3 << 25);       // pad_amount:   3+1 = 4 dwords
    g1[1] = (int)((8192u & 0xFFFFu) << 16);      // tensor_dim0[15:0] (=8192)
    g1[2] = (int)((8192u >> 16) | (128u << 16)); // tensor_dim0[31:16] | tensor_dim1[15:0]
    g1[3] = (int)(32u << 16);                    // tensor_dim1[31:16] | tile_dim0 = 32
    g1[4] = 128;                                 // tile_dim1 = 128, tile_dim2 = 0
    g1[5] = 8192;                                // tensor_dim0_stride[31:0]
    g1[6] = 0;                                   // stride hi | tensor_dim1_stride lo
    g1[7] = 0;
    i32x4 z4 = {0, 0, 0, 0};
#if __clang_major__ >= 23
    i32x8 z8 = {0, 0, 0, 0, 0, 0, 0, 0};
    __builtin_amdgcn_tensor_load_to_lds(g0, g1, z4, z4, z8, 0);
#else
    __builtin_amdgcn_tensor_load_to_lds(g0, g1, z4, z4, 0);
#endif
}
#endif

// out[b, fo, g] = sum_{fi,h} x[b,fi,h] * kernel[fo,fi,pt[h,g]] + bias[fo]
// == GEMM C(2048x8192) = X(2048x8192) * W(8192x8192),
// W[k=fi*256+h][n=fo*256+g] = kernel[fo][fi][pt[h][g]] gathered on the fly.
__global__ __launch_bounds__(256, 2)
void equi_group_gemm_wmma(const float* __restrict__ x,
                          const float* __restrict__ kern,
                          const float* __restrict__ bias,
                          const int*   __restrict__ pt,
                          float*       __restrict__ out)
{
    __shared__ float kfo[kFIn][kNS];        // kernel[fo,:,:]   (32 KB)
    __shared__ int   ptt[kBGN][kPPad];      // pt block, transposed [g][h_local]
    __shared__ float xt[2][kBM][kXPad];     // double-buffered x chunk

    const int tid   = threadIdx.x;
    const int lane  = tid & 31;
    const int wave  = tid >> 5;
    const int wrow  = wave >> 2;            // 0..1 : 64-row band
    const int wcol  = wave & 3;             // 0..3 : 16-col band
    const int lanem = lane & 15;
    const int laneh = lane >> 4;            // 0/1 (wave32 half select)

    const int b0 = blockIdx.x * kBM;
    const int fo = blockIdx.y;
    const int g0 = blockIdx.z * kBGN;

    const float* xbase = x + (size_t)b0 * kRowK;

    // ---- stage kernel[fo] slice into LDS (2048 float4 over 256 threads) ----
    {
        const v4f* src = (const v4f*)(kern + (size_t)fo * kFIn * kNS);
        v4f* dst = (v4f*)&kfo[0][0];
#pragma unroll
        for (int i = 0; i < 8; ++i)
            dst[tid + i * 256] = src[tid + i * 256];
    }

#if USE_TDM
    if (wave == 0) {  // kick off chunk 0 (fi=0, hb=0) into buffer 0
        tdm_load_x_tile(xbase, (unsigned)(uintptr_t)&xt[0][0][0]);
    }
#endif

    v8f acc[4] = {};  // four 16x16 f32 accumulators: 64 rows x 16 cols per wave

    for (int c = 0; c < kNChunks; ++c) {
        const int fi  = c & (kFIn - 1);
        const int hb  = (c >> 5) * kBH;
        const int buf = c & 1;

#if USE_TDM
        if (wave == 0) __builtin_amdgcn_s_wait_tensorcnt((short)0);  // chunk c landed
        __syncthreads();  // xt[buf] visible to all; compute(c-1) fully done
        if (wave == 0 && (c + 1) < kNChunks) {  // overlap: DMA chunk c+1 while computing c
            const int nfi = (c + 1) & (kFIn - 1);
            const int nhb = ((c + 1) >> 5) * kBH;
            tdm_load_x_tile(xbase + (size_t)nfi * kNS + nhb,
                            (unsigned)(uintptr_t)&xt[(c + 1) & 1][0][0]);
        }
#else
        __syncthreads();  // previous compute done reading xt[buf]
        {   // register-bounce staging fallback: 2 threads per row, b128 loads
            int r  = tid >> 1;
            int c0 = (tid & 1) * 16;
            const v4f* src = (const v4f*)(xbase + (size_t)r * kRowK
                                          + (size_t)fi * kNS + hb + c0);
            v4f t0 = src[0], t1 = src[1], t2 = src[2], t3 = src[3];
            v4f* dst = (v4f*)&xt[buf][r][c0];
            dst[0] = t0; dst[1] = t1; dst[2] = t2; dst[3] = t3;
        }
#endif

        if ((c & (kFIn - 1)) == 0) {
            // new h-block: stage pt[hb:hb+32, g0:g0+64] transposed (coalesced in g)
            for (int i = tid; i < kBGN * kBH; i += 256) {
                int k = i >> 6;
                int g = i & (kBGN - 1);
                ptt[g][k] = pt[(size_t)(hb + k) * kNS + (g0 + g)];
            }
            __syncthreads();
        }
#if !USE_TDM
        else {
            __syncthreads();  // xt[buf] stores visible
        }
#endif

        const float* krow = &kfo[fi][0];
        const float (*xb)[kXPad] = xt[buf];
#pragma unroll
        for (int ks = 0; ks < 8; ++ks) {
            const int kb = ks * 4 + laneh * 2;   // k-local of v0 (v1 = kb+1)
            // A fragments: 4 M-slices of this wave's 64-row band
            v2f a0 = *(const v2f*)&xb[wrow * 64 +  0 + lanem][kb];
            v2f a1 = *(const v2f*)&xb[wrow * 64 + 16 + lanem][kb];
            v2f a2 = *(const v2f*)&xb[wrow * 64 + 32 + lanem][kb];
            v2f a3 = *(const v2f*)&xb[wrow * 64 + 48 + lanem][kb];
            // one B fragment, gathered into registers; amortized over 4 WMMAs
            const int n = wcol * 16 + lanem;
            int2 s = *(const int2*)&ptt[n][kb];
            v2f bf; bf.x = krow[s.x]; bf.y = krow[s.y];

            acc[0] = __builtin_amdgcn_wmma_f32_16x16x4_f32(
                false, a0, false, bf, (short)0, acc[0], false, false);
            acc[1] = __builtin_amdgcn_wmma_f32_16x16x4_f32(
                false, a1, false, bf, (short)0, acc[1], false, false);
            acc[2] = __builtin_amdgcn_wmma_f32_16x16x4_f32(
                false, a2, false, bf, (short)0, acc[2], false, false);
            acc[3] = __builtin_amdgcn_wmma_f32_16x16x4_f32(
                false, a3, false, bf, (short)0, acc[3], false, false);
        }
    }

    // ---- epilogue: bias + store (C/D layout: VGPR r -> M = r + 8*laneh, N = lanem) ----
    const float bv = bias[fo];
    const int nidx = fo * kNS + g0 + wcol * 16 + lanem;
#pragma unroll
    for (int ms = 0; ms < 4; ++ms) {
        const int mbase = b0 + wrow * 64 + ms * 16 + laneh * 8;
        v8f cacc = acc[ms];
#pragma unroll
        for (int r = 0; r < 8; ++r)
            out[(size_t)(mbase + r) * kRowK + nidx] = cacc[r] + bv;
    }
}

extern "C" void kernel_launch(void* const* d_in, const int* in_sizes, int n_in,
                              void* d_out, int out_size, void* d_ws, size_t ws_size,
                              hipStream_t stream) {
    (void)in_sizes; (void)n_in; (void)out_size; (void)d_ws; (void)ws_size;
    const float* x    = (const float*)d_in[0];
    const float* kern = (const float*)d_in[1];
    const float* bias = (const float*)d_in[2];
    const int*   pt   = (const int*)d_in[3];
    float* out = (float*)d_out;

    dim3 grid(kB / kBM, kFOut, kNS / kBGN);  // 16 x 32 x 4 = 2048 workgroups
    equi_group_gemm_wmma<<<grid, dim3(256, 1, 1), 0, stream>>>(x, kern, bias, pt, out);
}